// ConvolutionLayers_2800318677025
// MI455X (gfx1250) — compile-verified
//
#include <hip/hip_runtime.h>
#include <hip/hip_bf16.h>

typedef __attribute__((ext_vector_type(2))) float v2f;
typedef __attribute__((ext_vector_type(8))) float v8f;

#define GCN_N   50000
#define GCN_E   800000
#define GCN_D   128
#define GCN_FFN 512
#define GCN_BOT 64
#define GCN_L   3

// ---------------------------------------------------------------------------
// fp32 WMMA GEMM: C[N x M] = A[N x K] @ B[K x M] (+ bias, optional relu)
// One wave computes a 64x16 output block (4 row-tiles x 1 col-tile) with
// V_WMMA_F32_16X16X4_F32. The B fragment is loaded once per K-step and reused
// across 4 independent accumulators (breaks the WMMA->WMMA RAW chain and
// amortizes weight loads 4x).
// A-layout (16x4, 2 VGPRs): lanes 0-15 hold K={k,k+1}, lanes 16-31 K={k+2,k+3}
// B-layout (4x16, 2 VGPRs): same K split, N = lane&15
// C/D layout (16x16, 8 VGPRs): N = lane&15, M = vgpr + 8*(lane>>4)
// ---------------------------------------------------------------------------
__global__ __launch_bounds__(128)
void gemm_f32_wmma(const float* __restrict__ A, const float* __restrict__ B,
                   const float* __restrict__ bias, float* __restrict__ C,
                   int N, int K, int M, int do_relu)
{
    const int lane   = threadIdx.x & 31;
    const int lo     = lane & 15;
    const int hi     = lane >> 4;
    const int tilesM = M >> 4;
    const int tilesN = (N + 15) >> 4;
    const int nGroup = (tilesN + 3) >> 2;          // groups of 4 row-tiles
    const int waveId = blockIdx.x * 4 + (threadIdx.x >> 5);
    const int group  = waveId / tilesM;
    const int tileCol = waveId - group * tilesM;
    if (group >= nGroup) return;                   // wave-uniform guard

    // Clamp trailing subtiles to the last valid row-tile (uniform; duplicated
    // subtiles recompute identical values and store them to identical
    // addresses -> deterministic).
    int tIdx[4];
#pragma unroll
    for (int t = 0; t < 4; ++t) {
        int ti = group * 4 + t;
        tIdx[t] = (ti < tilesN) ? ti : (tilesN - 1);
    }

    const int col = tileCol * 16 + lo;             // B/C column for this lane
    const float* Ap0 = A + (size_t)(tIdx[0] * 16 + lo) * K;
    const float* Ap1 = A + (size_t)(tIdx[1] * 16 + lo) * K;
    const float* Ap2 = A + (size_t)(tIdx[2] * 16 + lo) * K;
    const float* Ap3 = A + (size_t)(tIdx[3] * 16 + lo) * K;
    const float* Bc  = B + col;

    v8f acc0 = {}, acc1 = {}, acc2 = {}, acc3 = {};
#pragma unroll 4
    for (int k = 0; k < K; k += 4) {
        const int ka = k + 2 * hi;
        v2f b;
        b.x = Bc[(size_t)ka * M];
        b.y = Bc[(size_t)(ka + 1) * M];
        v2f a0 = *(const v2f*)(Ap0 + ka);
        v2f a1 = *(const v2f*)(Ap1 + ka);
        v2f a2 = *(const v2f*)(Ap2 + ka);
        v2f a3 = *(const v2f*)(Ap3 + ka);
        acc0 = __builtin_amdgcn_wmma_f32_16x16x4_f32(false, a0, false, b, (short)0, acc0, false, false);
        acc1 = __builtin_amdgcn_wmma_f32_16x16x4_f32(false, a1, false, b, (short)0, acc1, false, false);
        acc2 = __builtin_amdgcn_wmma_f32_16x16x4_f32(false, a2, false, b, (short)0, acc2, false, false);
        acc3 = __builtin_amdgcn_wmma_f32_16x16x4_f32(false, a3, false, b, (short)0, acc3, false, false);
    }

    const float bv = bias ? bias[col] : 0.0f;
    v8f accs[4] = {acc0, acc1, acc2, acc3};
#pragma unroll
    for (int t = 0; t < 4; ++t) {
        float* Cb = C + (size_t)(tIdx[t] * 16 + 8 * hi) * M + col;
#pragma unroll
        for (int r = 0; r < 8; ++r) {
            float v = accs[t][r] + bv;
            if (do_relu) v = fmaxf(v, 0.0f);
            Cb[(size_t)r * M] = v;
        }
    }
}

// ---------------------------------------------------------------------------
// Support kernels
// ---------------------------------------------------------------------------
__global__ void fill_zero_kernel(float* __restrict__ p, size_t n)
{
    size_t i = (size_t)blockIdx.x * blockDim.x + threadIdx.x;
    if (i < n) p[i] = 0.0f;
}

__global__ void degree_kernel(const long long* __restrict__ dst,
                              float* __restrict__ deg, int E)
{
    int e = blockIdx.x * blockDim.x + threadIdx.x;
    if (e < E) atomicAdd(&deg[(size_t)dst[e]], 1.0f);
}

__global__ void norm_kernel(float* __restrict__ deg_norm, int N)
{
    int i = blockIdx.x * blockDim.x + threadIdx.x;
    if (i < N) deg_norm[i] = rsqrtf(deg_norm[i] + 1.0f);
}

// One wave (32 lanes) per edge; each lane moves 4 of the 128 channels (float4).
__global__ void scatter_edges_kernel(const long long* __restrict__ src,
                                     const long long* __restrict__ dst,
                                     const float* __restrict__ xw,
                                     const float* __restrict__ nrm,
                                     float* __restrict__ agg, int E)
{
    int tid = blockIdx.x * blockDim.x + threadIdx.x;
    int e   = tid >> 5;
    if (e >= E) return;
    int lane = tid & 31;
    long long s = src[e], d = dst[e];
    float coef = nrm[s] * nrm[d];
    float4 v = *((const float4*)(xw + (size_t)s * GCN_D) + lane);
    float* out = agg + (size_t)d * GCN_D + lane * 4;
    atomicAdd(out + 0, v.x * coef);
    atomicAdd(out + 1, v.y * coef);
    atomicAdd(out + 2, v.z * coef);
    atomicAdd(out + 3, v.w * coef);
}

// out[i,:] = relu(agg[i,:] + xw[i,:]*norm[i]^2 + bias[:])  (out may alias agg)
__global__ void finalize_kernel(const float* __restrict__ agg,
                                const float* __restrict__ xw,
                                const float* __restrict__ nrm,
                                const float* __restrict__ bias,
                                float* __restrict__ out, int N)
{
    int tid  = blockIdx.x * blockDim.x + threadIdx.x;
    int node = tid >> 5;
    if (node >= N) return;
    int lane = tid & 31;
    float nn = nrm[node];
    nn = nn * nn;
    size_t base = (size_t)node * GCN_D + lane * 4;
    float4 a = *(const float4*)(agg + base);
    float4 x = *(const float4*)(xw + base);
    float4 b = *(const float4*)(bias + lane * 4);
    float4 o;
    o.x = fmaxf(a.x + x.x * nn + b.x, 0.0f);
    o.y = fmaxf(a.y + x.y * nn + b.y, 0.0f);
    o.z = fmaxf(a.z + x.z * nn + b.z, 0.0f);
    o.w = fmaxf(a.w + x.w * nn + b.w, 0.0f);
    *(float4*)(out + base) = o;
}

// ---------------------------------------------------------------------------
extern "C" void kernel_launch(void* const* d_in, const int* in_sizes, int n_in,
                              void* d_out, int out_size, void* d_ws, size_t ws_size,
                              hipStream_t stream)
{
    (void)in_sizes; (void)n_in; (void)out_size; (void)ws_size;

    const float*     x     = (const float*)d_in[0];
    const long long* ei    = (const long long*)d_in[1];   // int64 [2, E]
    const float*     convW = (const float*)d_in[2];       // [L, D, D]
    const float*     convB = (const float*)d_in[3];       // [L, D]
    const float*     fc1w  = (const float*)d_in[4];       // [D, FFN]
    const float*     fc1b  = (const float*)d_in[5];
    const float*     fc2w  = (const float*)d_in[6];       // [FFN, BOT]
    const float*     fc2b  = (const float*)d_in[7];
    float*           out   = (float*)d_out;               // [N, BOT]

    const long long* src = ei;
    const long long* dst = ei + GCN_E;

    // Carve workspace (256B aligned slices)
    auto align256 = [](size_t v) { return (v + 255) & ~(size_t)255; };
    char* ws = (char*)d_ws;
    float* nrm = (float*)ws;  ws += align256((size_t)GCN_N * sizeof(float));
    float* xw  = (float*)ws;  ws += align256((size_t)GCN_N * GCN_D * sizeof(float));
    float* agg = (float*)ws;  ws += align256((size_t)GCN_N * GCN_D * sizeof(float));
    float* h1  = (float*)ws;  ws += align256((size_t)GCN_N * GCN_FFN * sizeof(float));

    const int tilesN = (GCN_N + 15) / 16;        // 3125
    const int nGroup = (tilesN + 3) / 4;         // 782 groups of 64 rows

    // ---- degree + symmetric normalization (shared by all layers) ----
    fill_zero_kernel<<<(GCN_N + 255) / 256, 256, 0, stream>>>(nrm, (size_t)GCN_N);
    degree_kernel<<<(GCN_E + 255) / 256, 256, 0, stream>>>(dst, nrm, GCN_E);
    norm_kernel<<<(GCN_N + 255) / 256, 256, 0, stream>>>(nrm, GCN_N);

    const size_t nd = (size_t)GCN_N * GCN_D;
    const float* xin = x;
    for (int l = 0; l < GCN_L; ++l) {
        // xw = xin @ W_l   (fp32 WMMA)
        int waves = nGroup * (GCN_D / 16);
        gemm_f32_wmma<<<(waves + 3) / 4, 128, 0, stream>>>(
            xin, convW + (size_t)l * GCN_D * GCN_D, nullptr, xw,
            GCN_N, GCN_D, GCN_D, 0);
        // agg = 0; agg[dst] += coef * xw[src]
        fill_zero_kernel<<<(int)((nd + 255) / 256), 256, 0, stream>>>(agg, nd);
        scatter_edges_kernel<<<(GCN_E * 32 + 255) / 256, 256, 0, stream>>>(
            src, dst, xw, nrm, agg, GCN_E);
        // agg = relu(agg + xw*norm^2 + b_l)
        finalize_kernel<<<(GCN_N * 32 + 255) / 256, 256, 0, stream>>>(
            agg, xw, nrm, convB + (size_t)l * GCN_D, agg, GCN_N);
        xin = agg;
    }

    // h1 = relu(xin @ fc1w + fc1b)
    {
        int waves = nGroup * (GCN_FFN / 16);
        gemm_f32_wmma<<<(waves + 3) / 4, 128, 0, stream>>>(
            xin, fc1w, fc1b, h1, GCN_N, GCN_D, GCN_FFN, 1);
    }
    // out = h1 @ fc2w + fc2b
    {
        int waves = nGroup * (GCN_BOT / 16);
        gemm_f32_wmma<<<(waves + 3) / 4, 128, 0, stream>>>(
            h1, fc2w, fc2b, out, GCN_N, GCN_FFN, GCN_BOT, 0);
    }
}